// Lowpass_87479893885733
// MI455X (gfx1250) — compile-verified
//
#include <hip/hip_runtime.h>

// out[b,c] = P @ x[b,c] @ P^T per 32x32 image, P = Di[:, :15] @ D[:15, :].
// Two fp32 GEMMs per image via V_WMMA_F32_16X16X4_F32 (wave32, 16x16 tiles, K in chunks of 4).

typedef __attribute__((ext_vector_type(2))) float v2f;
typedef __attribute__((ext_vector_type(4))) float v4f;
typedef __attribute__((ext_vector_type(8))) float v8f;

#define IMG_N 32
#define WAVES_PER_WG 8

__device__ __forceinline__ v8f wmma_f32(v2f a, v2f b, v8f c) {
    // 8 args: (neg_a, A, neg_b, B, c_mod, C, reuse_a, reuse_b)
    return __builtin_amdgcn_wmma_f32_16x16x4_f32(false, a, false, b, (short)0, c,
                                                 false, false);
}

// Build the 32x32 projection matrix P into global scratch.
// P[h][w] = (1 + 2*sum_{k=1..14} cos(pi*(2h+1)k/64) cos(pi*(2w+1)k/64)) / 32
__global__ void build_P(float* __restrict__ P) {
    int idx = blockIdx.x * blockDim.x + threadIdx.x;
    if (idx >= IMG_N * IMG_N) return;
    int h = idx >> 5, w = idx & 31;
    const double PI = 3.14159265358979323846;
    double acc = 0.0;
    for (int k = 1; k < 15; ++k) {
        acc += cos(PI * (2 * h + 1) * k / 64.0) * cos(PI * (2 * w + 1) * k / 64.0);
    }
    P[idx] = (float)((1.0 + 2.0 * acc) / 32.0);
}

__global__ __launch_bounds__(32 * WAVES_PER_WG)
void lowpass_dct(const float* __restrict__ x, float* __restrict__ out,
                 const float* __restrict__ Pg, int nImages) {
    __shared__ float sP[IMG_N * IMG_N];                    // shared projection matrix
    __shared__ float sBuf[WAVES_PER_WG][IMG_N * IMG_N];    // per-wave image / temp

    const int tid  = threadIdx.x;
    const int lane = tid & 31;          // wave32
    const int wave = tid >> 5;

    // Cooperative load of P: 1024 floats = 256 float4, one per thread.
    ((v4f*)sP)[tid] = ((const v4f*)Pg)[tid];
    __syncthreads();   // before any divergence

    const int img = blockIdx.x * WAVES_PER_WG + wave;
    if (img >= nImages) return;        // uniform per wave -> EXEC stays all-ones

    float* sX = sBuf[wave];

    // Stage image into LDS, coalesced float4 loads: 8 x (32 lanes x 16B).
    {
        const v4f* src = (const v4f*)(x + (size_t)img * (IMG_N * IMG_N));
        v4f*       dst = (v4f*)sX;
#pragma unroll
        for (int t = 0; t < 8; ++t) dst[t * 32 + lane] = src[t * 32 + lane];
    }

    const int cl    = lane & 15;        // column within 16-wide tile / row select
    const int khalf = (lane >> 4) * 2;  // K sub-pair select (ISA 7.12.2 fp32 layouts)
    const int rhalf = (lane >> 4) * 8;  // C/D row-half select

    // ---- Stage 1: T = P @ x -------------------------------------------------
    v8f T00 = {}, T01 = {}, T10 = {}, T11 = {};
#pragma unroll
    for (int kk = 0; kk < 8; ++kk) {    // K = 32 in chunks of 4
        const int kb = 4 * kk + khalf;
        // A (16x4 of P): lane m=cl holds A[m, kb], A[m, kb+1]
        v2f a0, a1, b0, b1;
        a0.x = sP[cl * IMG_N + kb];          a0.y = sP[cl * IMG_N + kb + 1];
        a1.x = sP[(16 + cl) * IMG_N + kb];   a1.y = sP[(16 + cl) * IMG_N + kb + 1];
        // B (4x16 of x): vgpr0 = row kb, vgpr1 = row kb+1, column = cl
        b0.x = sX[kb * IMG_N + cl];          b0.y = sX[(kb + 1) * IMG_N + cl];
        b1.x = sX[kb * IMG_N + 16 + cl];     b1.y = sX[(kb + 1) * IMG_N + 16 + cl];
        T00 = wmma_f32(a0, b0, T00);
        T01 = wmma_f32(a0, b1, T01);
        T10 = wmma_f32(a1, b0, T10);
        T11 = wmma_f32(a1, b1, T11);
    }

    // Spill T over the image buffer (LDS ops are in-order within a wave).
#pragma unroll
    for (int v = 0; v < 8; ++v) {
        sX[(v + rhalf) * IMG_N + cl]             = T00[v];
        sX[(v + rhalf) * IMG_N + 16 + cl]        = T01[v];
        sX[(16 + v + rhalf) * IMG_N + cl]        = T10[v];
        sX[(16 + v + rhalf) * IMG_N + 16 + cl]   = T11[v];
    }

    // ---- Stage 2: out = T @ P^T  (B[k,n] = P[n_global, k_global]) ----------
    v8f O00 = {}, O01 = {}, O10 = {}, O11 = {};
#pragma unroll
    for (int kk = 0; kk < 8; ++kk) {
        const int kb = 4 * kk + khalf;
        v2f a0, a1, b0, b1;
        // A (16x4 of T): contiguous pair in row cl / 16+cl
        a0.x = sX[cl * IMG_N + kb];          a0.y = sX[cl * IMG_N + kb + 1];
        a1.x = sX[(16 + cl) * IMG_N + kb];   a1.y = sX[(16 + cl) * IMG_N + kb + 1];
        // B (4x16 of P^T): B[k, n] = P[n, k] -> contiguous pair in P row n
        b0.x = sP[cl * IMG_N + kb];          b0.y = sP[cl * IMG_N + kb + 1];
        b1.x = sP[(16 + cl) * IMG_N + kb];   b1.y = sP[(16 + cl) * IMG_N + kb + 1];
        O00 = wmma_f32(a0, b0, O00);
        O01 = wmma_f32(a0, b1, O01);
        O10 = wmma_f32(a1, b0, O10);
        O11 = wmma_f32(a1, b1, O11);
    }

    // Bounce result through LDS for coalesced float4 global stores.
#pragma unroll
    for (int v = 0; v < 8; ++v) {
        sX[(v + rhalf) * IMG_N + cl]             = O00[v];
        sX[(v + rhalf) * IMG_N + 16 + cl]        = O01[v];
        sX[(16 + v + rhalf) * IMG_N + cl]        = O10[v];
        sX[(16 + v + rhalf) * IMG_N + 16 + cl]   = O11[v];
    }
    {
        v4f*       dst = (v4f*)(out + (size_t)img * (IMG_N * IMG_N));
        const v4f* src = (const v4f*)sX;
#pragma unroll
        for (int t = 0; t < 8; ++t) dst[t * 32 + lane] = src[t * 32 + lane];
    }
}

extern "C" void kernel_launch(void* const* d_in, const int* in_sizes, int n_in,
                              void* d_out, int out_size, void* d_ws, size_t ws_size,
                              hipStream_t stream) {
    const float* x   = (const float*)d_in[0];
    float*       out = (float*)d_out;
    float*       P   = (float*)d_ws;   // 32*32 floats = 4 KB of scratch

    build_P<<<4, 256, 0, stream>>>(P);

    const int nImages = in_sizes[0] / (IMG_N * IMG_N);   // 8192*3 = 24576
    const int blocks  = (nImages + WAVES_PER_WG - 1) / WAVES_PER_WG;
    lowpass_dct<<<blocks, 32 * WAVES_PER_WG, 0, stream>>>(x, out, P, nImages);
}